// Roland_55731495633401
// MI455X (gfx1250) — compile-verified
//
#include <hip/hip_runtime.h>

#define DIN 128
#define HH 128

typedef float v2f __attribute__((ext_vector_type(2)));
typedef float v8f __attribute__((ext_vector_type(8)));

// ---------------- init (zero reduction buffers) ----------------
__global__ void k_init(float* __restrict__ acc, unsigned* __restrict__ emax,
                       float* __restrict__ denom, int N) {
  int i = blockIdx.x * blockDim.x + threadIdx.x;
  if (i < N * 128) acc[i] = 0.f;
  if (i < N) { emax[i] = 0u; denom[i] = 0.f; }
}

// ---------------- async global->LDS staging helpers (gfx1250) ----------------
__device__ __forceinline__ void async_g2l_b128(void* lds, const void* g) {
#if defined(__gfx1250__)
  unsigned ldsoff = (unsigned)(uintptr_t)lds;                 // low 32 bits of generic = LDS offset
  unsigned long long ga = (unsigned long long)(uintptr_t)g;
  asm volatile("global_load_async_to_lds_b128 %0, %1, off"
               :: "v"(ldsoff), "v"(ga) : "memory");
#else
  *(float4*)lds = *(const float4*)g;                          // host / non-gfx1250 pass
#endif
}
__device__ __forceinline__ void wait_asynccnt0() {
#if defined(__gfx1250__)
  asm volatile("s_wait_asynccnt 0x0" ::: "memory");
#endif
}

// ---------------- WMMA f32 GEMM: out[M,Nout] = A[M,K] @ W[Nout,K]^T + bias ----------------
// block = 256 threads (8 waves). The 16-row A tile is staged to LDS once per block with
// GLOBAL_LOAD_ASYNC_TO_LDS_B128 (ASYNCcnt) and consumed by all 8 waves via ds_load_b64.
// Wave w computes the 16x16 tile at rows [16*blockIdx.x,+16), cols [16*(8*blockIdx.y+w),+16).
// K-loop of V_WMMA_F32_16X16X4_F32, fragments per ISA 7.12.2:
//   A 16x4 : VGPR0 = K{k0, k0+2}, VGPR1 = K{k0+1, k0+3}; M = lane%16
//   B 4x16 : symmetric (K in VGPR/half, N = lane%16)
//   C 16x16: VGPR v -> M = v + 8*(lane/16), N = lane%16
#define A_STRIDE 132  // padded row stride (floats): ds_load_b64 fragment reads are bank-conflict-free
__global__ void __launch_bounds__(256)
k_gemm_wmma(const float* __restrict__ A, const float* __restrict__ W,
            const float* __restrict__ bias, float* __restrict__ out,
            int M, int K, int Nout, int do_relu) {
  __shared__ __align__(16) float As[16 * A_STRIDE];

  const int wave = threadIdx.x >> 5;
  const int lane = threadIdx.x & 31;
  const int half = lane >> 4;
  const int r    = lane & 15;
  const int m0 = blockIdx.x * 16;
  const int n0 = (blockIdx.y * 8 + wave) * 16;

  // stage A tile: 16 rows x K floats, 16B chunks
  const int kq = K >> 2;                 // chunks per row (K/4 floats of 16B)
  const int nchunk = 16 * kq;
  for (int cc = threadIdx.x; cc < nchunk; cc += 256) {
    int row = cc / kq, part = cc % kq;
    int ra = m0 + row; if (ra >= M) ra = M - 1;       // clamp (harmless duplicate)
    async_g2l_b128(&As[row * A_STRIDE + part * 4], A + (size_t)ra * K + part * 4);
  }
  wait_asynccnt0();
  __syncthreads();

  const int rowW = n0 + r;
  const float* pw = W + (size_t)rowW * K + 2 * half;
  const float* pa = &As[r * A_STRIDE + 2 * half];

  v8f c = {0.f, 0.f, 0.f, 0.f, 0.f, 0.f, 0.f, 0.f};
#pragma unroll 8
  for (int k0 = 0; k0 < K; k0 += 4) {
    float2 af = *(const float2*)(pa + k0);            // ds_load_b64 from staged tile
    v2f a, b;
    a.x = af.x; a.y = af.y;
    b.x = pw[k0]; b.y = pw[k0 + 1];
    c = __builtin_amdgcn_wmma_f32_16x16x4_f32(false, a, false, b, (short)0, c, false, false);
  }
#pragma unroll
  for (int v = 0; v < 8; ++v) {
    int row = m0 + v + 8 * half;
    int col = n0 + r;
    if (row < M) {
      float val = c[v] + bias[col];
      if (do_relu) val = fmaxf(val, 0.f);
      out[(size_t)row * Nout + col] = val;
    }
  }
}

// ---------------- order-preserving float<->uint for atomic segment-max ----------------
__device__ __forceinline__ unsigned f2ord(float f) {
  unsigned u = __float_as_uint(f);
  return (u & 0x80000000u) ? ~u : (u | 0x80000000u);
}
__device__ __forceinline__ float ord2f(unsigned k) {
  unsigned u = (k & 0x80000000u) ? (k & 0x7FFFFFFFu) : ~k;
  return __uint_as_float(u);
}

// ---------------- edge pass 1: e = att . leakyrelu(xl[src]+xr[dst]); segment max ----------------
__global__ void k_edge_logit(const long long* __restrict__ ei, const float* __restrict__ xl,
                             const float* __restrict__ xr, const float* __restrict__ att,
                             float* __restrict__ e, unsigned* __restrict__ emax,
                             long long E, long long EE) {
  long long eid = (long long)blockIdx.x * 8 + (threadIdx.x >> 5);
  if (eid >= EE) return;
  int lane = threadIdx.x & 31;
  long long s, d;
  if (eid < E) { s = ei[eid]; d = ei[E + eid]; } else { s = d = eid - E; }
  const float4* xl4 = (const float4*)(xl + s * 128);
  const float4* xr4 = (const float4*)(xr + d * 128);
  const float4* at4 = (const float4*)att;
  float4 a = xl4[lane], b = xr4[lane], w = at4[lane];
  float vx = a.x + b.x, vy = a.y + b.y, vz = a.z + b.z, vw = a.w + b.w;
  vx = vx > 0.f ? vx : 0.2f * vx;
  vy = vy > 0.f ? vy : 0.2f * vy;
  vz = vz > 0.f ? vz : 0.2f * vz;
  vw = vw > 0.f ? vw : 0.2f * vw;
  float p = vx * w.x + vy * w.y + vz * w.z + vw * w.w;
#pragma unroll
  for (int off = 16; off; off >>= 1) p += __shfl_xor(p, off, 32);
  if (lane == 0) {
    e[eid] = p;
    atomicMax(&emax[d], f2ord(p));
  }
}

// ---------------- edge pass 2: ez = exp(e - max[dst]); segment sum ----------------
__global__ void k_edge_soft(const long long* __restrict__ ei, const float* __restrict__ e,
                            const unsigned* __restrict__ emax, float* __restrict__ ez,
                            float* __restrict__ denom, long long E, long long EE) {
  long long eid = (long long)blockIdx.x * blockDim.x + threadIdx.x;
  if (eid >= EE) return;
  long long d = (eid < E) ? ei[E + eid] : (eid - E);
  float v = __expf(e[eid] - ord2f(emax[d]));
  ez[eid] = v;
  atomicAdd(&denom[d], v);
}

// ---------------- edge pass 3: acc[dst] += alpha * xl[src] ----------------
__global__ void k_edge_scatter(const long long* __restrict__ ei, const float* __restrict__ xl,
                               const float* __restrict__ ez, const float* __restrict__ denom,
                               float* __restrict__ acc, long long E, long long EE) {
  long long eid = (long long)blockIdx.x * 8 + (threadIdx.x >> 5);
  if (eid >= EE) return;
  int lane = threadIdx.x & 31;
  long long s, d;
  if (eid < E) { s = ei[eid]; d = ei[E + eid]; } else { s = d = eid - E; }
  float alpha = ez[eid] / denom[d];
  const float4* xl4 = (const float4*)(xl + s * 128);
  float4 v = xl4[lane];
  float* dst = acc + d * 128 + lane * 4;
  atomicAdd(dst + 0, v.x * alpha);
  atomicAdd(dst + 1, v.y * alpha);
  atomicAdd(dst + 2, v.z * alpha);
  atomicAdd(dst + 3, v.w * alpha);
}

// ---------------- bias + relu ----------------
__global__ void k_bias_relu(const float* __restrict__ acc, const float* __restrict__ bconv,
                            float* __restrict__ h1, int N) {
  int i = blockIdx.x * blockDim.x + threadIdx.x;
  if (i < N * 128) h1[i] = fmaxf(acc[i] + bconv[i & 127], 0.f);
}

// ---------------- pack W_hh [384,128] -> Wp[(k4*384 + t)*4 + j] for coalesced scan loads ----------------
__global__ void k_packW(const float* __restrict__ Whh, float* __restrict__ Wp) {
  int gid = blockIdx.x * blockDim.x + threadIdx.x;
  if (gid >= 384 * 128) return;
  int t = gid >> 7;
  int kk = gid & 127;
  int k4 = kk >> 2, j = kk & 3;
  Wp[(((size_t)k4 * 384 + t) << 2) + j] = Whh[gid];
}

// ---------------- sequential GRU scan: single persistent block, h in LDS ----------------
__global__ void __launch_bounds__(384)
k_gru_scan(const float* __restrict__ gi, const float* __restrict__ Wp,
           const float* __restrict__ b_hh, const float* __restrict__ h0,
           float* __restrict__ hseq, int T) {
  __shared__ __align__(16) float hs[128];
  __shared__ float gh[384];
  __shared__ float bh[384];
  const int t = threadIdx.x;  // 0..383
  if (t < 128) hs[t] = h0[t];
  bh[t] = b_hh[t];
  __syncthreads();
  const float4* Wp4 = (const float4*)Wp;
  const float4* hp4 = (const float4*)hs;
  for (int step = 0; step < T; ++step) {
    // gh = W_hh @ h + b_hh  (coalesced float4 loads from L2-resident packed W)
    float a = 0.f;
#pragma unroll 8
    for (int k4 = 0; k4 < 32; ++k4) {
      float4 w = Wp4[k4 * 384 + t];
      float4 hv = hp4[k4];
      a += w.x * hv.x + w.y * hv.y + w.z * hv.z + w.w * hv.w;
    }
    gh[t] = a + bh[t];
    __syncthreads();
    if (t < 128) {
      const float* g = gi + (size_t)step * 384;
      float rg = 1.f / (1.f + __expf(-(g[t] + gh[t])));
      float zg = 1.f / (1.f + __expf(-(g[128 + t] + gh[128 + t])));
      float ng = tanhf(g[256 + t] + rg * gh[256 + t]);
      float hn = (1.f - zg) * ng + zg * hs[t];
      hs[t] = hn;
      hseq[(size_t)step * 128 + t] = hn;
    }
    __syncthreads();
  }
}

// ---------------- final head: out[row] = h2[row] . Wlin2 + blin2 ----------------
__global__ void k_head(const float* __restrict__ h2, const float* __restrict__ W2,
                       const float* __restrict__ b2, float* __restrict__ out, int N) {
  int row = blockIdx.x * 8 + (threadIdx.x >> 5);
  if (row >= N) return;
  int lane = threadIdx.x & 31;
  const float4* h4 = (const float4*)(h2 + (size_t)row * 128);
  const float4* w4 = (const float4*)W2;
  float4 a = h4[lane], w = w4[lane];
  float p = a.x * w.x + a.y * w.y + a.z * w.z + a.w * w.w;
#pragma unroll
  for (int off = 16; off; off >>= 1) p += __shfl_xor(p, off, 32);
  if (lane == 0) out[row] = p + b2[0];
}

extern "C" void kernel_launch(void* const* d_in, const int* in_sizes, int n_in,
                              void* d_out, int out_size, void* d_ws, size_t ws_size,
                              hipStream_t stream) {
  (void)n_in; (void)out_size; (void)ws_size;
  const float*      x     = (const float*)d_in[0];
  const long long*  ei    = (const long long*)d_in[1];
  const float*      Wl    = (const float*)d_in[2];
  const float*      bl    = (const float*)d_in[3];
  const float*      Wr    = (const float*)d_in[4];
  const float*      br    = (const float*)d_in[5];
  const float*      att   = (const float*)d_in[6];
  const float*      bconv = (const float*)d_in[7];
  const float*      W_ih  = (const float*)d_in[8];
  const float*      W_hh  = (const float*)d_in[9];
  const float*      b_ih  = (const float*)d_in[10];
  const float*      b_hh  = (const float*)d_in[11];
  const float*      h0    = (const float*)d_in[12];
  const float*      Wlin  = (const float*)d_in[13];
  const float*      blin  = (const float*)d_in[14];
  const float*      Wlin2 = (const float*)d_in[15];
  const float*      blin2 = (const float*)d_in[16];
  float* out = (float*)d_out;

  const int N = in_sizes[0] / DIN;
  const long long E = in_sizes[1] / 2;
  const long long EE = E + N;

  // workspace carve-up (N*128 / N*384 chunks are 16-byte multiples; e/ez/emax/denom are not padded)
  float* p = (float*)d_ws;
  float* xl   = p; p += (size_t)N * 128;
  float* xr   = p; p += (size_t)N * 128;
  float* acc  = p; p += (size_t)N * 128;
  float* h1   = p; p += (size_t)N * 128;
  float* gi   = p; p += (size_t)N * 384;
  float* hseq = p; p += (size_t)N * 128;
  float* h2   = p; p += (size_t)N * 128;
  float* e    = p; p += (size_t)EE;
  float* ez   = p; p += (size_t)EE;
  unsigned* emax = (unsigned*)p; p += N;
  float* denom = p; p += N;
  float* Wp   = p; p += 384 * 128;

  const int mt = (N + 15) / 16;
  dim3 blk(256);

  k_init<<<(N * 128 + 255) / 256, blk, 0, stream>>>(acc, emax, denom, N);
  k_gemm_wmma<<<dim3(mt, 1), blk, 0, stream>>>(x, Wl, bl, xl, N, 128, 128, 0);
  k_gemm_wmma<<<dim3(mt, 1), blk, 0, stream>>>(x, Wr, br, xr, N, 128, 128, 0);
  k_edge_logit<<<(unsigned)((EE + 7) / 8), blk, 0, stream>>>(ei, xl, xr, att, e, emax, E, EE);
  k_edge_soft<<<(unsigned)((EE + 255) / 256), blk, 0, stream>>>(ei, e, emax, ez, denom, E, EE);
  k_edge_scatter<<<(unsigned)((EE + 7) / 8), blk, 0, stream>>>(ei, xl, ez, denom, acc, E, EE);
  k_bias_relu<<<(N * 128 + 255) / 256, blk, 0, stream>>>(acc, bconv, h1, N);
  k_gemm_wmma<<<dim3(mt, 3), blk, 0, stream>>>(h1, W_ih, b_ih, gi, N, 128, 384, 0);
  k_packW<<<(384 * 128 + 255) / 256, blk, 0, stream>>>(W_hh, Wp);
  k_gru_scan<<<1, 384, 0, stream>>>(gi, Wp, b_hh, h0, hseq, N);
  k_gemm_wmma<<<dim3(mt, 1), blk, 0, stream>>>(hseq, Wlin, blin, h2, N, 128, 128, 1);
  k_head<<<(N + 7) / 8, blk, 0, stream>>>(h2, Wlin2, blin2, out, N);
}